// SKA_GroupAgg_31404800868795
// MI455X (gfx1250) — compile-verified
//
#include <hip/hip_runtime.h>
#include <cstdint>
#include <cstddef>

// ---------------------------------------------------------------------------
// SKA group aggregation for MI455X (gfx1250).
//   xs[n,g,h,w]  = sum_{c in group} x[n, g*8+c, h, w]           (LDS tile)
//   og[n,g,h,w]  = sum_{kh,kw} w[n,g,kh*3+kw,h,w] * xs_pad[h+kh-1, w+kw-1]
//   out[n,oc]    = og[n, oc/8]                                   (broadcast x8)
//
// Pure bandwidth problem (~210 MB moved, ~53 MFLOP): design streams w once
// through async global->LDS DMA (ASYNCcnt-tracked, double buffered), stages
// the 9x-reused group-sum xs in LDS, and uses non-temporal 128-bit stores for
// the write-once broadcast output.
// ---------------------------------------------------------------------------

typedef float v4f __attribute__((ext_vector_type(4)));

#define NB      4            // batch
#define IC      64
#define HH      256
#define WW      256
#define GR      8            // groups
#define GIN     8            // input channels per group
#define GOUT    8            // output channels per group
#define TAPS    9            // 3x3
#define TILE_H  16           // output rows per block
#define CROWS   4            // rows per pipelined w-chunk
#define NCHUNK  (TILE_H / CROWS)
#define PLANE   (HH * WW)    // 65536 elements per channel plane
#define XS_ROWS (TILE_H + 2) // 18 (row halo)
#define XS_ELEM (XS_ROWS * WW)                 // 4608 floats
#define WB_ELEM (2 * TAPS * CROWS * WW)        // 18432 floats (double buffer)
#define SMEM_BYTES ((size_t)(XS_ELEM + WB_ELEM) * sizeof(float))  // 92160 B

// gfx1250 async global -> LDS copy (tracked by ASYNCcnt, no VGPR data path).
__device__ __forceinline__ void async_copy_b128(uint32_t lds_byte_addr,
                                                const void* sgpr_base,
                                                uint32_t voff_bytes) {
  asm volatile("global_load_async_to_lds_b128 %0, %1, %2"
               :
               : "v"(lds_byte_addr), "v"(voff_bytes), "s"(sgpr_base)
               : "memory");
}

template <int CNT>
__device__ __forceinline__ void wait_asynccnt() {
  asm volatile("s_wait_asynccnt %0" : : "i"(CNT) : "memory");
}

__global__ __launch_bounds__(256) void SKA_GroupAgg_gfx1250_kernel(
    const float* __restrict__ x,   // [N, IC, H, W]
    const float* __restrict__ w,   // [N, G, 9, H, W]
    float* __restrict__ out) {     // [N, OC, H, W]
  extern __shared__ float smem[];
  float* xs = smem;                // [18][256] group-sum tile (+/-1 row halo)
  float* wb = smem + XS_ELEM;      // [2][9][4][256] async w staging

  const int tid  = (int)threadIdx.x;
  const int rowc = tid >> 6;       // 0..3  : row within a 4-row chunk
  const int qc   = tid & 63;       // 0..63 : float4 column index
  const int r0   = (int)blockIdx.x * TILE_H;
  const int g    = (int)blockIdx.y;
  const int n    = (int)blockIdx.z;

  const float* xg = x   + (size_t)(n * IC + g * GIN) * PLANE;
  const float* wg = w   + (size_t)(n * GR + g) * TAPS * PLANE;
  float*       og = out + (size_t)(n * IC + g * GOUT) * PLANE;

  // ---- prologue: launch async DMA of w chunk 0 into LDS buffer 0 ----------
  {
    const int y = r0 + rowc;
#pragma unroll
    for (int tap = 0; tap < TAPS; ++tap) {
      const uint32_t lds = (uint32_t)(uintptr_t)
          &wb[((0 * TAPS + tap) * CROWS + rowc) * WW + 4 * qc];
      const uint32_t off = (uint32_t)((tap * PLANE + y * WW + 4 * qc) * 4);
      async_copy_b128(lds, (const void*)wg, off);
    }
  }

  // ---- phase 1: xs[row][:] = sum over the group's 8 channels --------------
  // Coalesced b128 loads; zero rows outside the image (zero padding).
#pragma unroll
  for (int j = 0; j < 5; ++j) {
    const int row = rowc + j * 4;          // 0..19, keep <18
    if (row < XS_ROWS) {
      const int y = r0 - 1 + row;
      v4f s = {0.0f, 0.0f, 0.0f, 0.0f};
      if (y >= 0 && y < HH) {
        const float* p = xg + (size_t)y * WW + 4 * qc;
#pragma unroll
        for (int c = 0; c < GIN; ++c) {
          s += *(const v4f*)(p + (size_t)c * PLANE);
        }
      }
      *(v4f*)&xs[row * WW + 4 * qc] = s;
    }
  }
  __syncthreads();  // xs tile visible to all waves

  // ---- main pipeline over 4-row chunks ------------------------------------
#pragma unroll
  for (int cb = 0; cb < NCHUNK; ++cb) {
    const int buf = cb & 1;

    // Prefetch next chunk's w via async DMA, then drain this chunk's copies.
    if (cb + 1 < NCHUNK) {
      const int nbuf = (cb + 1) & 1;
      const int y    = r0 + (cb + 1) * CROWS + rowc;
#pragma unroll
      for (int tap = 0; tap < TAPS; ++tap) {
        const uint32_t lds = (uint32_t)(uintptr_t)
            &wb[((nbuf * TAPS + tap) * CROWS + rowc) * WW + 4 * qc];
        const uint32_t off = (uint32_t)((tap * PLANE + y * WW + 4 * qc) * 4);
        async_copy_b128(lds, (const void*)wg, off);
      }
      wait_asynccnt<TAPS>();   // in-order: first 9 (= chunk cb) are complete
    } else {
      wait_asynccnt<0>();
    }
    __syncthreads();           // every wave's share of chunk cb is in LDS

    // 3x6 xs window for this thread's 4 output pixels (cols 4qc..4qc+3).
    const int i0 = cb * CROWS + rowc;      // xs row for kh = 0
    float win[3][6];
#pragma unroll
    for (int kh = 0; kh < 3; ++kh) {
      const float* xr = &xs[(i0 + kh) * WW];
      const v4f m = *(const v4f*)(xr + 4 * qc);     // aligned b128
      win[kh][0] = (qc > 0)  ? xr[4 * qc - 1] : 0.0f;  // left zero-pad
      win[kh][1] = m.x; win[kh][2] = m.y;
      win[kh][3] = m.z; win[kh][4] = m.w;
      win[kh][5] = (qc < 63) ? xr[4 * qc + 4] : 0.0f;  // right zero-pad
    }

    v4f acc = {0.0f, 0.0f, 0.0f, 0.0f};
#pragma unroll
    for (int kh = 0; kh < 3; ++kh) {
#pragma unroll
      for (int kw = 0; kw < 3; ++kw) {
        const int tap = kh * 3 + kw;
        const v4f wv = *(const v4f*)
            &wb[((buf * TAPS + tap) * CROWS + rowc) * WW + 4 * qc];
        acc.x += wv.x * win[kh][kw + 0];
        acc.y += wv.y * win[kh][kw + 1];
        acc.z += wv.z * win[kh][kw + 2];
        acc.w += wv.w * win[kh][kw + 3];
      }
    }

    // Broadcast to the group's 8 output channels: write-once -> NT stores.
    const int y = r0 + cb * CROWS + rowc;
    float* ob = og + (size_t)y * WW + 4 * qc;
#pragma unroll
    for (int oc = 0; oc < GOUT; ++oc) {
      __builtin_nontemporal_store(acc, (v4f*)(ob + (size_t)oc * PLANE));
    }
    __syncthreads();  // all readers done before buf is overwritten next iter
  }
}

extern "C" void kernel_launch(void* const* d_in, const int* in_sizes, int n_in,
                              void* d_out, int out_size, void* d_ws,
                              size_t ws_size, hipStream_t stream) {
  (void)in_sizes; (void)n_in; (void)out_size; (void)d_ws; (void)ws_size;
  const float* x = (const float*)d_in[0];
  const float* w = (const float*)d_in[1];
  // d_in[2] = out_channels (fixed 64 per the reference shapes) -> hardcoded.
  float* out = (float*)d_out;

  dim3 grid(HH / TILE_H, GR, NB);  // (16, 8, 4) = 512 blocks
  dim3 block(256);                 // 8 wave32s
  SKA_GroupAgg_gfx1250_kernel<<<grid, block, SMEM_BYTES, stream>>>(x, w, out);
}